// IdentifyModel_72490458022035
// MI455X (gfx1250) — compile-verified
//
#include <hip/hip_runtime.h>

typedef __attribute__((ext_vector_type(2))) float v2f;
typedef __attribute__((ext_vector_type(8))) float v8f;

#define N_REL 3

// ---------------------------------------------------------------------------
// Embedding gather: X[i, 0:4] = size_emb[x_size[i]], X[i, 4:36] = code_emb[x_code[i]]
// ---------------------------------------------------------------------------
__global__ void embed_kernel(const int* __restrict__ x_code,
                             const int* __restrict__ x_size,
                             const float* __restrict__ size_emb,
                             const float* __restrict__ code_emb,
                             float* __restrict__ X, int n) {
    long t = (long)blockIdx.x * blockDim.x + threadIdx.x;
    long total = (long)n * 36;
    if (t >= total) return;
    int i = (int)(t / 36);
    int f = (int)(t % 36);
    X[t] = (f < 4) ? size_emb[x_size[i] * 4 + f]
                   : code_emb[x_code[i] * 32 + (f - 4)];
}

__global__ void zero_kernel(float* __restrict__ p, long n) {
    long t = (long)blockIdx.x * blockDim.x + threadIdx.x;
    if (t < n) p[t] = 0.0f;
}

// Per-(relation, node) in-degree counts, then inverted (layer invariant).
__global__ void count_kernel(const int* __restrict__ dst,
                             const int* __restrict__ etype,
                             float* __restrict__ cnt, int n_edges, int n_nodes) {
    int t = blockIdx.x * blockDim.x + threadIdx.x;
    if (t >= n_edges) return;
    atomicAdd(&cnt[(long)etype[t] * n_nodes + dst[t]], 1.0f);
}

__global__ void inv_kernel(const float* __restrict__ cnt,
                           float* __restrict__ invc, long n) {
    long t = (long)blockIdx.x * blockDim.x + threadIdx.x;
    if (t < n) invc[t] = 1.0f / fmaxf(cnt[t], 1.0f);
}

// ---------------------------------------------------------------------------
// WMMA dense transform over FULL 16-row tiles:
//   Y[16T x DOUT] = act(X)[16T x DIN] @ W[DIN x DOUT] (+bias)
// f32 16x16x4 layout (ISA 7.12.2):
//   A: lane l -> M = l&15, khalf = l>>4; a[v] = X[M][k0 + 2*khalf + v]
//   B: lane l -> N = l&15;               b[v] = W[k0 + 2*khalf + v][N]
//   C/D: c[j] -> (M = j + 8*khalf, N = l&15)
// W staged in LDS with padded column stride 16*NTILES (pad columns = 0), so
// B reads need no guards; A loaded unconditionally as 8B v2f (full tiles only).
// ---------------------------------------------------------------------------
template <int DIN, int DOUT, bool RELU_IN, bool HAS_BIAS>
__global__ __launch_bounds__(256) void wmma_transform_kernel(
    const float* __restrict__ X, const float* __restrict__ W,
    const float* __restrict__ bias, float* __restrict__ Y, int ntiles_m) {
    constexpr int KSTEPS = DIN / 4;             // DIN in {36,24,16,8}: exact
    constexpr int NTILES = (DOUT + 15) / 16;
    constexpr int LDW    = 16 * NTILES;         // padded LDS row stride

    __shared__ float lW[DIN * LDW];
    for (int i = threadIdx.x; i < DIN * LDW; i += blockDim.x) {
        const int k = i / LDW;
        const int n = i % LDW;
        lW[i] = (n < DOUT) ? W[k * DOUT + n] : 0.0f;
    }
    __syncthreads();

    const int lane   = threadIdx.x & 31;
    const int wv     = threadIdx.x >> 5;
    const int khalf  = lane >> 4;
    const int nl     = lane & 15;
    const int nwaves = (int)((gridDim.x * blockDim.x) >> 5);
    const int wid    = (int)(blockIdx.x * (blockDim.x >> 5) + wv);

    for (int mt = wid; mt < ntiles_m; mt += nwaves) {   // wave-uniform loop
        const long row0 = (long)mt * 16;
        const float* __restrict__ xrow = X + (row0 + nl) * DIN;

        // Preload the A tile (reused across all N-tiles)
        v2f a[KSTEPS];
#pragma unroll
        for (int ks = 0; ks < KSTEPS; ++ks) {
            v2f av = *(const v2f*)(xrow + ks * 4 + 2 * khalf);
            if (RELU_IN) { av[0] = fmaxf(av[0], 0.0f); av[1] = fmaxf(av[1], 0.0f); }
            a[ks] = av;
        }

#pragma unroll
        for (int nt = 0; nt < NTILES; ++nt) {
            const int ncol = nt * 16 + nl;
            float binit = 0.0f;
            if (HAS_BIAS) binit = bias[ncol < DOUT ? ncol : DOUT - 1]; // clamped, OOB cols unstored
            v8f c;
#pragma unroll
            for (int j = 0; j < 8; ++j) c[j] = binit;

#pragma unroll
            for (int ks = 0; ks < KSTEPS; ++ks) {
                const int ka = ks * 4 + 2 * khalf;
                v2f b;
                b[0] = lW[ka * LDW + nt * 16 + nl];
                b[1] = lW[(ka + 1) * LDW + nt * 16 + nl];
                c = __builtin_amdgcn_wmma_f32_16x16x4_f32(
                        false, a[ks], false, b, (short)0, c, false, false);
            }

#pragma unroll
            for (int j = 0; j < 8; ++j) {
                if (NTILES * 16 == DOUT || ncol < DOUT)  // compile-time when exact
                    Y[(row0 + j + 8 * khalf) * DOUT + ncol] = c[j];
            }
        }
    }
}

// Generic scalar tail for the (N % 16) remainder rows (not launched when N%16==0).
__global__ void tail_transform_kernel(const float* __restrict__ X,
                                      const float* __restrict__ W,
                                      const float* __restrict__ bias,
                                      float* __restrict__ Y,
                                      long row_start, long n_nodes,
                                      int din, int dout, int relu_in) {
    long t = (long)blockIdx.x * blockDim.x + threadIdx.x;
    long total = (n_nodes - row_start) * dout;
    if (t >= total) return;
    const long i = row_start + t / dout;
    const int  n = (int)(t % dout);
    float acc = bias ? bias[n] : 0.0f;
    for (int k = 0; k < din; ++k) {
        float x = X[i * din + k];
        if (relu_in) x = fmaxf(x, 0.0f);
        acc += x * W[k * dout + n];
    }
    Y[i * dout + n] = acc;
}

// ---------------------------------------------------------------------------
// Edge scatter for one relation: Out[dst] += Y[src] * invc[dst]
// Sub-warp group of G lanes per edge (G = pow2 >= DOUT), lane-in-group = feature.
// ---------------------------------------------------------------------------
template <int DOUT>
__global__ void scatter_kernel(const float* __restrict__ Y,
                               const int* __restrict__ src,
                               const int* __restrict__ dst,
                               const int* __restrict__ etype,
                               const float* __restrict__ invc,
                               float* __restrict__ Out,
                               int n_edges, int rel) {
    constexpr int G = ((DOUT & (DOUT - 1)) != 0 || DOUT > 32) ? 32 : DOUT;
    const long tid     = (long)blockIdx.x * blockDim.x + threadIdx.x;
    const int  f       = (int)(tid % G);
    const long group   = tid / G;
    const long ngroups = ((long)gridDim.x * blockDim.x) / G;
    for (long e = group; e < n_edges; e += ngroups) {
        if (etype[e] != rel) continue;
        const int s = src[e];
        const int d = dst[e];
        const float ic = invc[d];
        if (G == DOUT || f < DOUT) {
            atomicAdd(&Out[(long)d * DOUT + f], Y[(long)s * DOUT + f] * ic);
        }
    }
}

// ---------------------------------------------------------------------------
// Final head with fused ReLU on input: out[i,j] = relu(X[i,:]) @ lin_w + lin_b
// ---------------------------------------------------------------------------
__global__ void final_linear_kernel(const float* __restrict__ X,
                                    const float* __restrict__ lw,
                                    const float* __restrict__ lb,
                                    float* __restrict__ out, int n) {
    long t = (long)blockIdx.x * blockDim.x + threadIdx.x;
    if (t >= (long)n * 2) return;
    const long i = t >> 1;
    const int  j = (int)(t & 1);
    float acc = lb[j];
#pragma unroll
    for (int d = 0; d < 4; ++d) acc += fmaxf(X[i * 4 + d], 0.0f) * lw[d * 2 + j];
    out[t] = acc;
}

// ---------------------------------------------------------------------------
template <int DIN, int DOUT, bool RELU_IN>
static void run_layer(const float* xin, float* xout, float* Yb,
                      const float* w, const float* root, const float* bias,
                      const float* invc, const int* src, const int* dst,
                      const int* etype, int N, int E, hipStream_t stream) {
    const int BLK      = 256;
    const int mt_full  = N / 16;
    const long rem0    = (long)mt_full * 16;
    const int rem      = N - (int)rem0;
    const int wgrid    = 2048;
    const int sgrid    = 4096;

    // out = relu?(x) @ root + b
    wmma_transform_kernel<DIN, DOUT, RELU_IN, true><<<wgrid, BLK, 0, stream>>>(
        xin, root, bias, xout, mt_full);
    if (rem > 0)
        tail_transform_kernel<<<(rem * DOUT + BLK - 1) / BLK, BLK, 0, stream>>>(
            xin, root, bias, xout, rem0, N, DIN, DOUT, RELU_IN ? 1 : 0);

    for (int r = 0; r < N_REL; ++r) {
        const float* wr = w + (size_t)r * DIN * DOUT;
        wmma_transform_kernel<DIN, DOUT, RELU_IN, false><<<wgrid, BLK, 0, stream>>>(
            xin, wr, nullptr, Yb, mt_full);
        if (rem > 0)
            tail_transform_kernel<<<(rem * DOUT + BLK - 1) / BLK, BLK, 0, stream>>>(
                xin, wr, nullptr, Yb, rem0, N, DIN, DOUT, RELU_IN ? 1 : 0);
        scatter_kernel<DOUT><<<sgrid, BLK, 0, stream>>>(
            Yb, src, dst, etype, invc + (size_t)r * N, xout, E, r);
    }
}

extern "C" void kernel_launch(void* const* d_in, const int* in_sizes, int n_in,
                              void* d_out, int out_size, void* d_ws, size_t ws_size,
                              hipStream_t stream) {
    const int*   x_code    = (const int*)d_in[0];
    const int*   x_size    = (const int*)d_in[1];
    const int*   edge_idx  = (const int*)d_in[2];
    const int*   edge_type = (const int*)d_in[3];
    const float* size_emb  = (const float*)d_in[4];
    const float* code_emb  = (const float*)d_in[5];
    const float* w0 = (const float*)d_in[6],  *r0 = (const float*)d_in[7],  *b0 = (const float*)d_in[8];
    const float* w1 = (const float*)d_in[9],  *r1 = (const float*)d_in[10], *b1 = (const float*)d_in[11];
    const float* w2 = (const float*)d_in[12], *r2 = (const float*)d_in[13], *b2 = (const float*)d_in[14];
    const float* w3 = (const float*)d_in[15], *r3 = (const float*)d_in[16], *b3 = (const float*)d_in[17];
    const float* lin_w = (const float*)d_in[18];
    const float* lin_b = (const float*)d_in[19];

    const int N = in_sizes[0];
    const int E = in_sizes[3];
    const int* src = edge_idx;        // edge_index[0, :]
    const int* dst = edge_idx + E;    // edge_index[1, :]

    // Workspace (floats): X0(36N) | X1(24N) | Y(24N, doubles as cnt) | invc(3N)
    float* X0   = (float*)d_ws;
    float* X1   = X0 + (size_t)36 * N;
    float* Yb   = X1 + (size_t)24 * N;
    float* invc = Yb + (size_t)24 * N;
    float* cnt  = Yb;

    const int BLK = 256;

    // 1) input embedding
    {
        long total = (long)N * 36;
        embed_kernel<<<(unsigned)((total + BLK - 1) / BLK), BLK, 0, stream>>>(
            x_code, x_size, size_emb, code_emb, X0, N);
    }

    // 2) per-relation inverse in-degree (layer invariant)
    {
        long cn = (long)N_REL * N;
        zero_kernel<<<(unsigned)((cn + BLK - 1) / BLK), BLK, 0, stream>>>(cnt, cn);
        count_kernel<<<(E + BLK - 1) / BLK, BLK, 0, stream>>>(dst, edge_type, cnt, E, N);
        inv_kernel<<<(unsigned)((cn + BLK - 1) / BLK), BLK, 0, stream>>>(cnt, invc, cn);
    }

    // 3) RGCN layers; ReLU is fused into the NEXT consumer's input reads.
    run_layer<36, 24, false>(X0, X1, Yb, w0, r0, b0, invc, src, dst, edge_type, N, E, stream);
    run_layer<24, 16, true >(X1, X0, Yb, w1, r1, b1, invc, src, dst, edge_type, N, E, stream);
    run_layer<16,  8, true >(X0, X1, Yb, w2, r2, b2, invc, src, dst, edge_type, N, E, stream);
    run_layer< 8,  4, true >(X1, X0, Yb, w3, r3, b3, invc, src, dst, edge_type, N, E, stream);

    // 4) final head (applies the layer-3 ReLU on read)
    final_linear_kernel<<<(unsigned)(((long)N * 2 + BLK - 1) / BLK), BLK, 0, stream>>>(
        X0, lin_w, lin_b, (float*)d_out, N);
}